// Cace_58291296141968
// MI455X (gfx1250) — compile-verified
//
#include <hip/hip_runtime.h>
#include <hip/hip_bf16.h>
#include <math.h>

#define NN 4000        // nodes
#define NE 48000       // edges
#define NRB 8          // radial basis / transformed radial
#define NL 20          // angular monomials (l<=3)
#define NC 9           // channels
#define RIC (NRB*NL*NC)   // 1440 per-node feature count
#define CUT 5.5f
#define MPN 0.31622776601683794f  // 1/sqrt(10)

typedef __attribute__((ext_vector_type(2))) float v2f;
typedef __attribute__((ext_vector_type(8))) float v8f;

// l-list ordering: l=0; l=1:(100)(010)(001); l=2:(200)(110)(101)(020)(011)(002);
// l=3:(300)(210)(201)(120)(111)(102)(030)(021)(012)(003)
constexpr int LXc[NL]   = {0, 1,0,0, 2,1,1,0,0,0, 3,2,2,1,1,1,0,0,0,0};
constexpr int LYc[NL]   = {0, 0,1,0, 0,1,0,2,1,0, 0,1,0,2,1,0,3,2,1,0};
constexpr int LZc[NL]   = {0, 0,0,1, 0,0,1,0,1,2, 0,0,1,0,1,2,0,1,2,3};
constexpr int LOFc[NL]  = {0, 1,1,1, 2,2,2,2,2,2, 3,3,3,3,3,3,3,3,3,3};
constexpr float PREFc[NL] = {1.f, 1.f,1.f,1.f, 1.f,2.f,2.f,1.f,2.f,1.f,
                             1.f,3.f,3.f,3.f,6.f,3.f,1.f,3.f,3.f,1.f};

__device__ __forceinline__ int l_of(int i) {
  return (i == 0) ? 0 : (i < 4) ? 1 : (i < 10) ? 2 : 3;
}

__global__ void k_zero(float* __restrict__ p, int n) {
  int i = blockIdx.x * blockDim.x + threadIdx.x;
  if (i < n) p[i] = 0.f;
}

__global__ void k_embed(const float* __restrict__ W_emb, const int* __restrict__ zn,
                        float* __restrict__ emb) {
  int n = blockIdx.x * blockDim.x + threadIdx.x;
  if (n >= NN) return;
  int z = zn[n];
  int idx = (z == 1) ? 0 : (z == 6) ? 1 : (z == 7) ? 2 : 3;
  emb[n * 3 + 0] = W_emb[idx * 3 + 0];
  emb[n * 3 + 1] = W_emb[idx * 3 + 1];
  emb[n * 3 + 2] = W_emb[idx * 3 + 2];
}

// Per-edge: bessel-RBF * poly cutoff (8), angular monomials (20), channel enc (9)
__global__ void k_edge_geom(const float* __restrict__ pos, const float* __restrict__ shifts,
                            const int* __restrict__ ei, const float* __restrict__ emb,
                            float* __restrict__ rc, float* __restrict__ ang,
                            float* __restrict__ enc) {
  int e = blockIdx.x * blockDim.x + threadIdx.x;
  if (e >= NE) return;
  int s = ei[e], t = ei[NE + e];
  float vx = pos[t * 3 + 0] - pos[s * 3 + 0] + shifts[e * 3 + 0];
  float vy = pos[t * 3 + 1] - pos[s * 3 + 1] + shifts[e * 3 + 1];
  float vz = pos[t * 3 + 2] - pos[s * 3 + 2] + shifts[e * 3 + 2];
  float r = sqrtf(vx * vx + vy * vy + vz * vz);
  float inv = 1.f / r;
  // poly cutoff p=6: 1 - 28u^6 + 48u^7 - 21u^8 for u<1
  float u = r * (1.f / CUT);
  float u2 = u * u, u4 = u2 * u2, u6 = u4 * u2;
  float f = (u < 1.f) ? (1.f - 28.f * u6 + 48.f * u6 * u - 21.f * u4 * u4) : 0.f;
  float pref = sqrtf(2.f / CUT) * inv * f;
  float w = 3.14159265358979323846f * r * (1.f / CUT);
#pragma unroll
  for (int n = 1; n <= NRB; n++) rc[e * NRB + (n - 1)] = pref * sinf((float)n * w);

  float x = vx * inv, y = vy * inv, z = vz * inv;
  float px[4] = {1.f, x, x * x, x * x * x};
  float py[4] = {1.f, y, y * y, y * y * y};
  float pz[4] = {1.f, z, z * z, z * z * z};
#pragma unroll
  for (int i = 0; i < NL; i++)
    ang[e * NL + i] = px[LXc[i]] * py[LYc[i]] * pz[LZc[i]];

  float es0 = emb[s * 3 + 0], es1 = emb[s * 3 + 1], es2 = emb[s * 3 + 2];
  float et0 = emb[t * 3 + 0], et1 = emb[t * 3 + 1], et2 = emb[t * 3 + 2];
  float esv[3] = {es0, es1, es2}, etv[3] = {et0, et1, et2};
#pragma unroll
  for (int a = 0; a < 3; a++)
#pragma unroll
    for (int b = 0; b < 3; b++) enc[e * NC + a * 3 + b] = esv[a] * etv[b];
}

// Scatter rc ⊗ ang ⊗ encv into acc[receiver]: thread per (edge, r, i), 9 atomics each
__global__ void k_scatter_ri(const float* __restrict__ rc, const float* __restrict__ ang,
                             const float* __restrict__ encv, const int* __restrict__ recv,
                             float* __restrict__ acc) {
  int tid = blockIdx.x * blockDim.x + threadIdx.x;
  if (tid >= NE * NRB * NL) return;
  int e = tid / (NRB * NL), ri = tid % (NRB * NL);
  int r = ri / NL, i = ri % NL;
  float v = rc[e * NRB + r] * ang[e * NL + i];
  float* dst = acc + (size_t)recv[e] * RIC + r * (NL * NC) + i * NC;
  const float* en = encv + e * NC;
#pragma unroll
  for (int c = 0; c < NC; c++) atomicAdd(dst + c, v * en[c]);
}

// msg_A scatter: S2[recv] += A[sender][s,i,c] * rc[e,s]
__global__ void k_scatter_msgA(const float* __restrict__ rc, const float* __restrict__ A,
                               const int* __restrict__ ei, float* __restrict__ S2) {
  int tid = blockIdx.x * blockDim.x + threadIdx.x;
  if (tid >= NE * NRB * NL) return;
  int e = tid / (NRB * NL), si = tid % (NRB * NL);
  int s = si / NL, i = si % NL;
  int snd = ei[e], t = ei[NE + e];
  float v = rc[e * NRB + s];
  const float* src = A + (size_t)snd * RIC + s * (NL * NC) + i * NC;
  float* dst = S2 + (size_t)t * RIC + s * (NL * NC) + i * NC;
#pragma unroll
  for (int c = 0; c < NC; c++) atomicAdd(dst + c, v * src[c]);
}

// Radial transform via V_WMMA_F32_16X16X4_F32 (two chained K=4 steps, K=8 total).
// Each wave: tile of 16 nodes x fixed i; fully unrolled c=0..8 (18 WMMAs/wave).
//   out[n,s,i,c] = sum_r S1[n,r,i,c] * W_rt[l(i),r,s]
// COMBINE: A[idx] = (out + S2[idx])*MP_NORM + A[idx]*W_nm[s, l(i), c]
template <bool COMBINE>
__global__ void k_radial_wmma(const float* __restrict__ S1, const float* __restrict__ W_rt,
                              float* __restrict__ A, const float* __restrict__ S2,
                              const float* __restrict__ W_nm) {
  int wave = (blockIdx.x * blockDim.x + threadIdx.x) >> 5;
  int lane = threadIdx.x & 31;
  int nb = wave / NL;       // node block (0..249)
  int i = wave % NL;        // angular index
  if (nb >= NN / 16) return;  // wave-uniform
  int m = lane & 15, half = lane >> 4;
  int l = l_of(i);
  const float* W = W_rt + l * (NRB * NRB);

  // B operand (4x16, K rows striped over lane halves); N-cols >=8 are zero pad
  v2f b0 = {}, b1 = {};
  if (m < NRB) {
    b0.x = W[(2 * half + 0) * NRB + m];      // k = {0..3}
    b0.y = W[(2 * half + 1) * NRB + m];
    b1.x = W[(4 + 2 * half + 0) * NRB + m];  // k = {4..7}
    b1.y = W[(4 + 2 * half + 1) * NRB + m];
  }

  int node_in = nb * 16 + m;
  const float* s1base = S1 + (size_t)node_in * RIC + i * NC;
  // store base: D-tile row M = v + 8*half maps to node nb*16 + v + 8*half, col s = m
  size_t obase = (size_t)(nb * 16 + 8 * half) * RIC + (size_t)m * (NL * NC) + (size_t)i * NC;
  float wnm = COMBINE ? W_nm[m * 36 + l * NC + (0)] : 0.f;  // re-read per c below
  (void)wnm;
  bool svalid = (m < NRB);

#pragma unroll
  for (int c = 0; c < NC; c++) {
    v2f a0, a1;  // A 16x4: lane m holds row m, k = 2*half + {0,1} (+4 second step)
    a0.x = s1base[(2 * half + 0) * (NL * NC) + c];
    a0.y = s1base[(2 * half + 1) * (NL * NC) + c];
    a1.x = s1base[(4 + 2 * half + 0) * (NL * NC) + c];
    a1.y = s1base[(4 + 2 * half + 1) * (NL * NC) + c];
    v8f acc = {};
    acc = __builtin_amdgcn_wmma_f32_16x16x4_f32(false, a0, false, b0, (short)0, acc,
                                                false, false);
    acc = __builtin_amdgcn_wmma_f32_16x16x4_f32(false, a1, false, b1, (short)0, acc,
                                                false, false);
    if (svalid) {  // output column s = m
      float wn = COMBINE ? W_nm[m * 36 + l * NC + c] : 0.f;
#pragma unroll
      for (int v = 0; v < 8; v++) {
        size_t idx = obase + (size_t)v * RIC + c;
        if (COMBINE) {
          A[idx] = (acc[v] + S2[idx]) * MPN + A[idx] * wn;
        } else {
          A[idx] = acc[v];
        }
      }
    }
  }
}

// B = symmetrize(A): k=0 -> A[i=0]; k=1..4 -> sum_i in l-group PREF*A^2.
// Writes feature slot `feat` of out (..., 2).
__global__ void k_symm(const float* __restrict__ A, float* __restrict__ out, int feat) {
  int tid = blockIdx.x * blockDim.x + threadIdx.x;
  if (tid >= NN * NRB) return;
  const float* a = A + (size_t)tid * (NL * NC);   // tid = n*8+s
  float* o = out + (size_t)tid * 5 * NC * 2 + feat;
  float accl[4][NC];
#pragma unroll
  for (int l = 0; l < 4; l++)
#pragma unroll
    for (int c = 0; c < NC; c++) accl[l][c] = 0.f;
#pragma unroll
  for (int i = 0; i < NL; i++) {
#pragma unroll
    for (int c = 0; c < NC; c++) {
      float v = a[i * NC + c];
      accl[LOFc[i]][c] += PREFc[i] * v * v;
    }
  }
#pragma unroll
  for (int c = 0; c < NC; c++) o[c * 2] = a[c];  // k=0
#pragma unroll
  for (int k = 1; k < 5; k++)
#pragma unroll
    for (int c = 0; c < NC; c++) o[(k * NC + c) * 2] = accl[k - 1][c];
}

// chi[n,c] = sum over s(8),k(5) of B (read from out feature 0)
__global__ void k_chi(const float* __restrict__ out, float* __restrict__ chi) {
  int tid = blockIdx.x * blockDim.x + threadIdx.x;
  if (tid >= NN * NC) return;
  int n = tid / NC, c = tid % NC;
  const float* base = out + ((size_t)n * NRB * 5 * NC + c) * 2;
  float acc = 0.f;
#pragma unroll
  for (int sk = 0; sk < NRB * 5; sk++) acc += base[sk * NC * 2];
  chi[tid] = acc;
}

// enc[e,c] *= chi[sender[e], c]   (builds per-edge weight for msg_B scatter)
__global__ void k_encx(float* __restrict__ enc, const float* __restrict__ chi,
                       const int* __restrict__ ei) {
  int tid = blockIdx.x * blockDim.x + threadIdx.x;
  if (tid >= NE * NC) return;
  int e = tid / NC, c = tid % NC;
  enc[tid] *= chi[ei[e] * NC + c];
}

extern "C" void kernel_launch(void* const* d_in, const int* in_sizes, int n_in,
                              void* d_out, int out_size, void* d_ws, size_t ws_size,
                              hipStream_t stream) {
  (void)in_sizes; (void)n_in; (void)out_size; (void)ws_size;
  const float* pos    = (const float*)d_in[0];
  const float* shifts = (const float*)d_in[1];
  const float* W_emb  = (const float*)d_in[2];
  const float* W_rt   = (const float*)d_in[3];
  const float* W_nm   = (const float*)d_in[4];
  const int*   zn     = (const int*)d_in[5];
  const int*   ei     = (const int*)d_in[6];
  float* out = (float*)d_out;

  float* ws = (float*)d_ws;
  size_t off = 0;
  float* emb  = ws + off; off += (size_t)NN * 3;
  float* rcb  = ws + off; off += (size_t)NE * NRB;
  float* angb = ws + off; off += (size_t)NE * NL;
  float* encb = ws + off; off += (size_t)NE * NC;
  float* S1   = ws + off; off += (size_t)NN * RIC;   // pre-RT accumulator (reused)
  float* Abuf = ws + off; off += (size_t)NN * RIC;   // A after radial transform
  float* S2   = ws + off; off += (size_t)NN * RIC;   // msg_A accumulator
  float* chi  = ws + off; off += (size_t)NN * NC;

  const int B = 256;
  const int nA = NN * RIC;
  const int nScat = NE * NRB * NL;
  const int wmmaBlocks = (NN / 16) * NL / 8;  // 5000 waves / 8 waves-per-block

  // ---- initial A ----
  k_zero<<<(nA + B - 1) / B, B, 0, stream>>>(S1, nA);
  k_embed<<<(NN + B - 1) / B, B, 0, stream>>>(W_emb, zn, emb);
  k_edge_geom<<<(NE + B - 1) / B, B, 0, stream>>>(pos, shifts, ei, emb, rcb, angb, encb);
  k_scatter_ri<<<(nScat + B - 1) / B, B, 0, stream>>>(rcb, angb, encb, ei + NE, S1);
  k_radial_wmma<false><<<wmmaBlocks, 256, 0, stream>>>(S1, W_rt, Abuf, nullptr, nullptr);
  k_symm<<<(NN * NRB + B - 1) / B, B, 0, stream>>>(Abuf, out, 0);
  k_chi<<<(NN * NC + B - 1) / B, B, 0, stream>>>(out, chi);

  // ---- one message-passing step ----
  k_zero<<<(nA + B - 1) / B, B, 0, stream>>>(S1, nA);
  k_zero<<<(nA + B - 1) / B, B, 0, stream>>>(S2, nA);
  k_encx<<<(NE * NC + B - 1) / B, B, 0, stream>>>(encb, chi, ei);
  k_scatter_ri<<<(nScat + B - 1) / B, B, 0, stream>>>(rcb, angb, encb, ei + NE, S1);
  k_scatter_msgA<<<(nScat + B - 1) / B, B, 0, stream>>>(rcb, Abuf, ei, S2);
  k_radial_wmma<true><<<wmmaBlocks, 256, 0, stream>>>(S1, W_rt, Abuf, S2, W_nm);
  k_symm<<<(NN * NRB + B - 1) / B, B, 0, stream>>>(Abuf, out, 1);
}